// CommunityDetectionModel_30331059044677
// MI455X (gfx1250) — compile-verified
//
#include <hip/hip_runtime.h>
#include <hip/hip_bf16.h>

// ---------------------------------------------------------------------------
// GCN pipeline for MI455X (gfx1250, wave32).
// GEMMs use V_WMMA_F32_16X16X4_F32 (full fp32 precision; K stepped by 4),
// with 2-way M-blocking per wave so each B fragment feeds two WMMAs.
// Edge propagation: one wave per edge, float atomics into L2-resident buffer.
// ---------------------------------------------------------------------------

typedef __attribute__((ext_vector_type(2))) float v2f;
typedef __attribute__((ext_vector_type(8))) float v8f;

#define LN_EPS 1e-5f

// ---------------------------------------------------------------------------
// WMMA GEMM: out[nrows x NOUT] = A[nrows x K] @ W[K x NOUT] (+ bias)
// One wave computes an MT*16-row stripe across all NT output tiles of 16 cols.
// A-frag layout (16x4 f32): lane = hi*16 + m;  holds A[m][k+2*hi], A[m][k+2*hi+1]
// B-frag layout (4x16 f32): lane = hi*16 + n;  holds W[k+2*hi][n], W[k+2*hi+1][n]
// C/D layout   (16x16 f32): vgpr v, lane: out[m0 + v + 8*hi][n = lane&15]
// nrows must be a multiple of MT*16 (100000 % 32 == 0).
// ---------------------------------------------------------------------------
template <int K, int NT, int MT>
__global__ __launch_bounds__(256)
void gemm_wmma_f32(const float* __restrict__ A,
                   const float* __restrict__ W,
                   const float* __restrict__ bias,
                   float* __restrict__ out,
                   int nrows) {
    constexpr int NOUT = NT * 16;
    const int wave = (blockIdx.x * blockDim.x + threadIdx.x) >> 5;
    const int lane = threadIdx.x & 31;
    const int m0 = wave * (MT * 16);
    if (m0 >= nrows) return;

    const int mr = lane & 15;   // row-in-tile for A, col-in-tile for B/D
    const int hi = lane >> 4;   // which K half-pair this lane carries

    v8f acc[MT][NT];
#pragma unroll
    for (int mt = 0; mt < MT; ++mt)
#pragma unroll
        for (int t = 0; t < NT; ++t)
            acc[mt][t] = (v8f){0.f, 0.f, 0.f, 0.f, 0.f, 0.f, 0.f, 0.f};

    const float* arow[MT];
#pragma unroll
    for (int mt = 0; mt < MT; ++mt)
        arow[mt] = A + (size_t)(m0 + mt * 16 + mr) * K;

#pragma unroll 2
    for (int k = 0; k < K; k += 4) {
        const int kb = k + hi * 2;
        v2f a[MT];
#pragma unroll
        for (int mt = 0; mt < MT; ++mt) {
            a[mt].x = arow[mt][kb];
            a[mt].y = arow[mt][kb + 1];
        }
        const float* w0 = W + (size_t)kb * NOUT;
        const float* w1 = w0 + NOUT;
#pragma unroll
        for (int t = 0; t < NT; ++t) {
            const int n = t * 16 + mr;
            v2f b;
            b.x = w0[n];
            b.y = w1[n];
#pragma unroll
            for (int mt = 0; mt < MT; ++mt)
                acc[mt][t] = __builtin_amdgcn_wmma_f32_16x16x4_f32(
                    false, a[mt], false, b, (short)0, acc[mt][t], false, false);
        }
    }

#pragma unroll
    for (int mt = 0; mt < MT; ++mt) {
#pragma unroll
        for (int t = 0; t < NT; ++t) {
            const int n = t * 16 + mr;
            const float bv = bias ? bias[n] : 0.f;
#pragma unroll
            for (int v = 0; v < 8; ++v) {
                const int m = m0 + mt * 16 + v + hi * 8;
                out[(size_t)m * NOUT + n] = acc[mt][t][v] + bv;
            }
        }
    }
}

// ---------------------------------------------------------------------------
// Degree / normalization
// ---------------------------------------------------------------------------
__global__ void deg_init_kernel(float* deg, int N) {
    int i = blockIdx.x * blockDim.x + threadIdx.x;
    if (i < N) deg[i] = 1.0f;  // self-loop
}

__global__ void deg_count_kernel(const int* __restrict__ col, float* deg, int E) {
    int e = blockIdx.x * blockDim.x + threadIdx.x;
    if (e < E) atomicAdd(&deg[col[e]], 1.0f);
}

__global__ void deg_inv_kernel(const float* __restrict__ deg, float* dinv, int N) {
    int i = blockIdx.x * blockDim.x + threadIdx.x;
    if (i < N) dinv[i] = rsqrtf(deg[i]);
}

// ---------------------------------------------------------------------------
// Self-loop contribution initializes accumulator: out[i] = hw[i] * dinv[i]^2
// ---------------------------------------------------------------------------
template <int F>
__global__ void init_self_kernel(const float* __restrict__ hw,
                                 const float* __restrict__ dinv,
                                 float* __restrict__ out, int N) {
    size_t i = (size_t)blockIdx.x * blockDim.x + threadIdx.x;
    if (i >= (size_t)N * F) return;
    const int node = (int)(i / F);
    const float dv = dinv[node];
    out[i] = hw[i] * dv * dv;
}

// ---------------------------------------------------------------------------
// Edge propagation: one wave per edge, lane carries F/32 contiguous floats.
// Gather from src[row], scale by dinv[row]*dinv[col], atomic-add to out[col].
// Feature array (<=51 MB) is L2-resident on MI455X (192 MB L2).
// ---------------------------------------------------------------------------
template <int F>
__global__ __launch_bounds__(256)
void scatter_edges_kernel(const float* __restrict__ src,
                          const int* __restrict__ rows,
                          const int* __restrict__ cols,
                          const float* __restrict__ dinv,
                          float* __restrict__ out, int E) {
    constexpr int PER = F / 32;
    const int wave = (blockIdx.x * blockDim.x + threadIdx.x) >> 5;
    const int lane = threadIdx.x & 31;
    if (wave >= E) return;
    const int r = rows[wave];
    const int c = cols[wave];
    const float nrm = dinv[r] * dinv[c];
    const float* s = src + (size_t)r * F + lane * PER;
    float* o = out + (size_t)c * F + lane * PER;
#pragma unroll
    for (int i = 0; i < PER; ++i)
        atomicAdd(o + i, s[i] * nrm);
}

// ---------------------------------------------------------------------------
// bias + LayerNorm + ReLU, in place. One wave per node, F=128, 4 feats/lane.
// ---------------------------------------------------------------------------
__global__ __launch_bounds__(256)
void post_ln_relu_kernel(float* __restrict__ h,
                         const float* __restrict__ b,
                         const float* __restrict__ g,
                         const float* __restrict__ be, int N) {
    const int wave = (blockIdx.x * blockDim.x + threadIdx.x) >> 5;
    const int lane = threadIdx.x & 31;
    if (wave >= N) return;

    float v[4];
    float sum = 0.f;
    const size_t base = (size_t)wave * 128 + lane * 4;
#pragma unroll
    for (int i = 0; i < 4; ++i) {
        v[i] = h[base + i] + b[lane * 4 + i];
        sum += v[i];
    }
#pragma unroll
    for (int off = 16; off > 0; off >>= 1) sum += __shfl_xor(sum, off, 32);
    const float mu = sum * (1.0f / 128.0f);

    float var = 0.f;
#pragma unroll
    for (int i = 0; i < 4; ++i) {
        const float d = v[i] - mu;
        var += d * d;
    }
#pragma unroll
    for (int off = 16; off > 0; off >>= 1) var += __shfl_xor(var, off, 32);
    var *= (1.0f / 128.0f);
    const float rs = rsqrtf(var + LN_EPS);

#pragma unroll
    for (int i = 0; i < 4; ++i) {
        const float y = (v[i] - mu) * rs * g[lane * 4 + i] + be[lane * 4 + i];
        h[base + i] = fmaxf(y, 0.f);
    }
}

// ---------------------------------------------------------------------------
// Elementwise bias add: h[i] += b[i % F]
// ---------------------------------------------------------------------------
template <int F>
__global__ void add_bias_kernel(float* __restrict__ h,
                                const float* __restrict__ b, int N) {
    size_t i = (size_t)blockIdx.x * blockDim.x + threadIdx.x;
    if (i >= (size_t)N * F) return;
    h[i] += b[i % F];
}

// ---------------------------------------------------------------------------
// Launcher
// ---------------------------------------------------------------------------
extern "C" void kernel_launch(void* const* d_in, const int* in_sizes, int n_in,
                              void* d_out, int out_size, void* d_ws, size_t ws_size,
                              hipStream_t stream) {
    constexpr int F = 128, H = 128, D = 64, C = 32;
    constexpr int MT = 2;  // 32 output rows per wave (100000 % 32 == 0)

    const float* x   = (const float*)d_in[0];
    const int*   ei  = (const int*)d_in[1];
    const float* W1  = (const float*)d_in[2];
    const float* b1  = (const float*)d_in[3];
    const float* g1  = (const float*)d_in[4];
    const float* be1 = (const float*)d_in[5];
    const float* W2  = (const float*)d_in[6];
    const float* b2  = (const float*)d_in[7];
    const float* Wc  = (const float*)d_in[8];
    const float* bc  = (const float*)d_in[9];
    float* logits    = (float*)d_out;

    const int N = in_sizes[0] / F;     // 100000
    const int E = in_sizes[1] / 2;     // 1600000
    const int* rows = ei;              // edge_index[0] = source
    const int* cols = ei + E;          // edge_index[1] = target

    // Workspace layout (with reuse; hw1 region is dead after propagate-1):
    //   [0, N*128)          hw1   -> later: agg2 (N*64) | hw2 (N*64)
    //   [N*128, N*256)      agg1 -> h2 (LN/ReLU in place)
    //   [N*256, N*257)      deg
    //   [N*257, N*258)      dinv
    float* ws   = (float*)d_ws;
    float* hw1  = ws;
    float* agg2 = ws;                      // aliases hw1 (first N*64)
    float* hw2  = ws + (size_t)N * 64;     // aliases hw1 (second N*64)
    float* agg1 = ws + (size_t)N * 128;    // becomes h2
    float* deg  = ws + (size_t)N * 256;
    float* dinv = deg + N;

    const int TPB = 256;
    const int wavesPerBlk = TPB / 32;
    auto cdiv = [](long long a, long long b) { return (int)((a + b - 1) / b); };

    // 1) degree + rsqrt (self-loops included via init=1)
    deg_init_kernel<<<cdiv(N, TPB), TPB, 0, stream>>>(deg, N);
    deg_count_kernel<<<cdiv(E, TPB), TPB, 0, stream>>>(cols, deg, E);
    deg_inv_kernel<<<cdiv(N, TPB), TPB, 0, stream>>>(deg, dinv, N);

    // 2) hw1 = x @ W1   (WMMA f32, 32 rows/wave)
    {
        const int nWaves = cdiv(N, 16 * MT);
        gemm_wmma_f32<F, H / 16, MT><<<cdiv(nWaves, wavesPerBlk), TPB, 0, stream>>>(
            x, W1, nullptr, hw1, N);
    }

    // 3) agg1 = selfloop(hw1) + scatter(edges)
    init_self_kernel<H><<<cdiv((long long)N * H, TPB), TPB, 0, stream>>>(hw1, dinv, agg1, N);
    scatter_edges_kernel<H><<<cdiv(E, wavesPerBlk), TPB, 0, stream>>>(
        hw1, rows, cols, dinv, agg1, E);

    // 4) h2 = relu(layernorm(agg1 + b1))   (in place)
    post_ln_relu_kernel<<<cdiv(N, wavesPerBlk), TPB, 0, stream>>>(agg1, b1, g1, be1, N);

    // 5) hw2 = h2 @ W2   (WMMA f32)
    {
        const int nWaves = cdiv(N, 16 * MT);
        gemm_wmma_f32<H, D / 16, MT><<<cdiv(nWaves, wavesPerBlk), TPB, 0, stream>>>(
            agg1, W2, nullptr, hw2, N);
    }

    // 6) agg2 = selfloop(hw2) + scatter(edges) ; then += b2
    init_self_kernel<D><<<cdiv((long long)N * D, TPB), TPB, 0, stream>>>(hw2, dinv, agg2, N);
    scatter_edges_kernel<D><<<cdiv(E, wavesPerBlk), TPB, 0, stream>>>(
        hw2, rows, cols, dinv, agg2, E);
    add_bias_kernel<D><<<cdiv((long long)N * D, TPB), TPB, 0, stream>>>(agg2, b2, N);

    // 7) logits = agg2 @ Wc + bc   (WMMA f32)
    {
        const int nWaves = cdiv(N, 16 * MT);
        gemm_wmma_f32<D, C / 16, MT><<<cdiv(nWaves, wavesPerBlk), TPB, 0, stream>>>(
            agg2, Wc, bc, logits, N);
    }
}